// voxelization_27771258536247
// MI455X (gfx1250) — compile-verified
//
#include <hip/hip_runtime.h>
#include <stdint.h>
#include <math.h>

#define NCLS 20

// ------------------------------------------------------------------
// hdr[0] = current N, hdr[1] = out cursor, hdr[2] = V (scan total)
// ------------------------------------------------------------------

__global__ void k_init_hdr(unsigned* hdr, unsigned N0) {
  hdr[0] = N0; hdr[1] = 0u; hdr[2] = 0u;
}

__global__ void k_fill0_u32(unsigned* __restrict__ p, unsigned n) {
  unsigned i = blockIdx.x * 256u + threadIdx.x;
  if (i < n) p[i] = 0u;
}

// zero exactly V*NCLS u16 counters (V read from device header)
__global__ void k_fill_counts(unsigned* __restrict__ cw, const unsigned* __restrict__ hdr) {
  unsigned nw = (hdr[2] * NCLS + 1u) >> 1;   // u32 words holding packed u16 pairs
  unsigned i = blockIdx.x * 256u + threadIdx.x;
  if (i < nw) cw[i] = 0u;
}

// ---- A: quantize points, write full_coors, set occupancy bit ----
__global__ void k_points(const float* __restrict__ pc, const int* __restrict__ bidx,
                         unsigned* __restrict__ vkeys, unsigned* __restrict__ flags,
                         float* __restrict__ out, const unsigned* __restrict__ hdr,
                         unsigned dx, unsigned dy, unsigned dz,
                         float fdx, float fdy, float fdz)
{
  unsigned p = blockIdx.x * 256u + threadIdx.x;
  unsigned N = hdr[0];
  if (p >= N) return;
  unsigned off = hdr[1];
  float x = pc[3*(size_t)p+0];
  float y = pc[3*(size_t)p+1];
  float z = pc[3*(size_t)p+2];
  int bb = bidx[p];
  // floor(sshape * (x - lo) / (hi - lo)) in f32, same op order as reference
  int xi = (int)floorf(fdx * (x - (-51.2f)) / 102.4f);
  int yi = (int)floorf(fdy * (y - (-51.2f)) / 102.4f);
  int zi = (int)floorf(fdz * (z - (-4.0f)) / 6.0f);
  unsigned vkey = (((unsigned)bb * dx + (unsigned)xi) * dy + (unsigned)yi) * dz + (unsigned)zi;
  vkeys[p] = vkey;
  atomicOr(&flags[vkey >> 5], 1u << (vkey & 31u));   // 4MB bitmask, L2-resident
  float* fc = out + (size_t)off + 4u * (size_t)p;
  fc[0] = (float)bb; fc[1] = (float)xi; fc[2] = (float)yi; fc[3] = (float)zi;
}

// ---- popcounts of bitmask words -> scan source ----
__global__ void k_popc(const unsigned* __restrict__ flags, unsigned* __restrict__ wscan, unsigned W)
{
  unsigned i = blockIdx.x * 256u + threadIdx.x;
  if (i < W) wscan[i] = (unsigned)__popc(flags[i]);
}

// ---- scan level kernel: 2048 u32 per block, tile staged with CDNA5
//      async global->LDS ops, exclusive scan written back in place ----
__global__ void k_scan_blocks(unsigned* __restrict__ data, unsigned* __restrict__ partials)
{
  __shared__ unsigned tile[2048];
  __shared__ unsigned ssum[256];
  unsigned t = threadIdx.x;
  size_t base = (size_t)blockIdx.x * 2048u;

  // async-copy this block's 8KB tile into LDS (2 x b128 per lane)
  {
    const unsigned* g0 = data + base + (size_t)t * 8u;
    unsigned l0 = (unsigned)(size_t)&tile[t * 8u];
    asm volatile("global_load_async_to_lds_b128 %0, %1, off"
                 :: "v"(l0), "v"(g0) : "memory");
    const unsigned* g1 = g0 + 4;
    unsigned l1 = l0 + 16u;
    asm volatile("global_load_async_to_lds_b128 %0, %1, off"
                 :: "v"(l1), "v"(g1) : "memory");
  }
  asm volatile("s_wait_asynccnt 0" ::: "memory");
  __syncthreads();

  unsigned v[8]; unsigned s = 0u;
#pragma unroll
  for (int i = 0; i < 8; ++i) { v[i] = tile[t * 8u + i]; s += v[i]; }
  ssum[t] = s;
  __syncthreads();
  for (unsigned o = 1u; o < 256u; o <<= 1) {
    unsigned q = (t >= o) ? ssum[t - o] : 0u;
    __syncthreads();
    ssum[t] += q;
    __syncthreads();
  }
  unsigned excl = ssum[t] - s;
#pragma unroll
  for (int i = 0; i < 8; ++i) { unsigned vv = v[i]; data[base + t * 8u + i] = excl; excl += vv; }
  if (t == 255u) partials[blockIdx.x] = ssum[255];
}

__global__ void k_scan_single(unsigned* __restrict__ data, unsigned n, unsigned* __restrict__ total)
{
  __shared__ unsigned ssum[256];
  unsigned t = threadIdx.x;
  unsigned v[8]; unsigned s = 0u;
#pragma unroll
  for (int i = 0; i < 8; ++i) {
    unsigned idx = t * 8u + i;
    v[i] = (idx < n) ? data[idx] : 0u;
    s += v[i];
  }
  ssum[t] = s;
  __syncthreads();
  for (unsigned o = 1u; o < 256u; o <<= 1) {
    unsigned q = (t >= o) ? ssum[t - o] : 0u;
    __syncthreads();
    ssum[t] += q;
    __syncthreads();
  }
  unsigned excl = ssum[t] - s;
#pragma unroll
  for (int i = 0; i < 8; ++i) {
    unsigned idx = t * 8u + i;
    if (idx < n) data[idx] = excl;
    excl += v[i];
  }
  if (t == 255u && total) *total = ssum[255];
}

__global__ void k_add_offsets(unsigned* __restrict__ data, const unsigned* __restrict__ partials)
{
  size_t i = (size_t)blockIdx.x * 256u + threadIdx.x;
  data[i] += partials[i >> 11];   // n is always a multiple of 2048 here
}

// ---- B: inverse indices + packed-u16 per-voxel label histogram ----
__global__ void k_inv_counts(const unsigned* __restrict__ vkeys, const unsigned* __restrict__ flags,
                             const unsigned* __restrict__ wscan, const int* __restrict__ lab,
                             unsigned* __restrict__ cwords, float* __restrict__ out,
                             const unsigned* __restrict__ hdr)
{
  unsigned p = blockIdx.x * 256u + threadIdx.x;
  unsigned N = hdr[0];
  if (p >= N) return;
  unsigned off = hdr[1];
  unsigned vkey = vkeys[p];
  unsigned w = vkey >> 5, bit = vkey & 31u;
  unsigned r = wscan[w] + (unsigned)__popc(flags[w] & ((1u << bit) - 1u));
  out[(size_t)off + 4u * (size_t)N + p] = (float)r;
  unsigned pidx = r * NCLS + (unsigned)lab[p];
  atomicAdd(&cwords[pidx >> 1], 1u << (16u * (pidx & 1u)));
}

// ---- C: per bitmask word, walk set bits -> coors, majority label, next inputs ----
__global__ void k_voxels(const unsigned* __restrict__ flags, const unsigned* __restrict__ wscan,
                         const unsigned short* __restrict__ counts,
                         const unsigned* __restrict__ hdr, float* __restrict__ out,
                         float* __restrict__ pc_next, int* __restrict__ b_next, int* __restrict__ lab_next,
                         unsigned W, unsigned dx, unsigned dy, unsigned dz,
                         float ivx, float ivy, float ivz)
{
  unsigned w = blockIdx.x * 256u + threadIdx.x;
  if (w >= W) return;
  unsigned bits = flags[w];
  if (!bits) return;
  unsigned N = hdr[0], off = hdr[1], V = hdr[2];
  unsigned r = wscan[w];
  size_t base = (size_t)off + 5u * (size_t)N;
  while (bits) {
    unsigned l = (unsigned)__ffs(bits) - 1u;
    bits &= bits - 1u;
    unsigned vc = w * 32u + l;
    const unsigned short* c = counts + (size_t)r * NCLS;
    unsigned best = 0u; int bl = 0;
#pragma unroll
    for (int k = 0; k < NCLS; ++k) {
      unsigned cv = c[k];
      if (cv > best) { best = cv; bl = k; }   // tie -> smallest label (matches ref)
    }
    unsigned zi = vc % dz; unsigned tt = vc / dz;
    unsigned yi = tt % dy; tt /= dy;
    unsigned xi = tt % dx; unsigned bb = tt / dx;
    float* cr = out + base + 4u * (size_t)r;
    cr[0] = (float)bb; cr[1] = (float)zi; cr[2] = (float)yi; cr[3] = (float)xi;
    out[base + 4u * (size_t)V + r] = (float)bl;
    pc_next[3u * (size_t)r + 0] = (float)xi * ivx + (-51.2f);
    pc_next[3u * (size_t)r + 1] = (float)yi * ivy + (-51.2f);
    pc_next[3u * (size_t)r + 2] = (float)zi * ivz + (-4.0f);
    b_next[r]   = (int)bb;
    lab_next[r] = bl;
    ++r;
  }
}

__global__ void k_update(unsigned* hdr)
{
  unsigned N = hdr[0], off = hdr[1], V = hdr[2];
  hdr[1] = off + 5u * N + 5u * V;
  hdr[0] = V;
}

// ------------------------------------------------------------------
// host side
// ------------------------------------------------------------------

static void scan_u32(unsigned* data, unsigned n, unsigned* tmp1, unsigned* tmp2,
                     unsigned* d_total, hipStream_t stream)
{
  if (n <= 2048u) {
    k_scan_single<<<1, 256, 0, stream>>>(data, n, d_total);
    return;
  }
  unsigned nb = n / 2048u;   // n is a multiple of 2048 by construction
  k_scan_blocks<<<nb, 256, 0, stream>>>(data, tmp1);
  scan_u32(tmp1, nb, tmp2, nullptr, d_total, stream);
  k_add_offsets<<<n / 256u, 256, 0, stream>>>(data, tmp1);
}

extern "C" void kernel_launch(void* const* d_in, const int* in_sizes, int n_in,
                              void* d_out, int out_size, void* d_ws, size_t ws_size,
                              hipStream_t stream)
{
  const float* points = (const float*)d_in[0];
  const int*   batch  = (const int*)d_in[1];
  const int*   labels = (const int*)d_in[2];
  float* out = (float*)d_out;

  const unsigned N0   = (unsigned)in_sizes[1];      // number of points
  const unsigned SMAX = 4u * 512u * 512u * 32u;     // scale-1 voxel space (2^25)
  const unsigned WMAX = SMAX / 32u;                 // bitmask words (1M)

  char* ws = (char*)d_ws;
  size_t o = 0;
  auto carve = [&](size_t bytes) -> void* {
    void* p = ws + o;
    o = (o + bytes + 255u) & ~(size_t)255u;
    return p;
  };
  unsigned* hdr    = (unsigned*)carve(256);
  unsigned* vkeys  = (unsigned*)carve((size_t)N0 * 4u);
  unsigned* flags  = (unsigned*)carve((size_t)WMAX * 4u);   // occupancy bitmask
  unsigned* wscan  = (unsigned*)carve((size_t)WMAX * 4u);   // popcount prefix sums
  unsigned* partA  = (unsigned*)carve(2048u * 4u);
  unsigned* partB  = (unsigned*)carve(2048u * 4u);
  size_t countsCap = ((size_t)((N0 < SMAX) ? N0 : SMAX) * NCLS + 1u) / 2u;  // u32 words of u16 pairs
  unsigned* counts = (unsigned*)carve(countsCap * 4u);
  float* pcA = (float*)carve((size_t)N0 * 3u * 4u);
  float* pcB = (float*)carve((size_t)N0 * 3u * 4u);
  int* bA   = (int*)carve((size_t)N0 * 4u);
  int* bB   = (int*)carve((size_t)N0 * 4u);
  int* labA = (int*)carve((size_t)N0 * 4u);
  int* labB = (int*)carve((size_t)N0 * 4u);
  (void)ws_size; (void)out_size; (void)n_in;

  k_init_hdr<<<1, 1, 0, stream>>>(hdr, N0);

  const int scales[5] = {1, 2, 4, 8, 16};
  const float* pcCur = points; const int* bCur = batch; const int* labCur = labels;
  unsigned Nbound = N0;

  for (int si = 0; si < 5; ++si) {
    int sc = scales[si];
    unsigned dx = 512u / (unsigned)sc, dy = 512u / (unsigned)sc, dz = 32u / (unsigned)sc;
    unsigned S = 4u * dx * dy * dz;               // power of two >= 8192
    unsigned W = S / 32u;                         // >= 256, multiple of 2048 when > 2048
    float fdx = (float)dx, fdy = (float)dy, fdz = (float)dz;
    float ivx = 102.4f / (512.0f / (float)sc);    // intervals, f32 like reference
    float ivy = 102.4f / (512.0f / (float)sc);
    float ivz = 6.0f   / (32.0f  / (float)sc);

    float* pcNxt  = (si & 1) ? pcB  : pcA;
    int*   bNxt   = (si & 1) ? bB   : bA;
    int*   labNxt = (si & 1) ? labB : labA;

    // 1) clear occupancy bitmask (<= 4MB)
    k_fill0_u32<<<(W + 255u) / 256u, 256, 0, stream>>>(flags, W);
    // 2) quantize + full_coors + occupancy bits
    k_points<<<(Nbound + 255u) / 256u, 256, 0, stream>>>(
        pcCur, bCur, vkeys, flags, out, hdr, dx, dy, dz, fdx, fdy, fdz);
    // 3) popcount words, exclusive scan -> ranks; total V -> hdr[2]
    k_popc<<<(W + 255u) / 256u, 256, 0, stream>>>(flags, wscan, W);
    scan_u32(wscan, W, partA, partB, hdr + 2, stream);
    // 4) zero exactly V*NCLS u16 counters (V device-resident after scan)
    {
      unsigned worstWords = (((Nbound < S) ? Nbound : S) * NCLS + 1u) / 2u;
      k_fill_counts<<<(worstWords + 255u) / 256u, 256, 0, stream>>>(counts, hdr);
    }
    // 5) inverse indices + label histogram
    k_inv_counts<<<(Nbound + 255u) / 256u, 256, 0, stream>>>(
        vkeys, flags, wscan, labCur, counts, out, hdr);
    // 6) per-voxel outputs + next-scale inputs
    k_voxels<<<(W + 255u) / 256u, 256, 0, stream>>>(
        flags, wscan, (const unsigned short*)counts, hdr, out,
        pcNxt, bNxt, labNxt, W, dx, dy, dz, ivx, ivy, ivz);
    // 7) advance N and output cursor
    k_update<<<1, 1, 0, stream>>>(hdr);

    pcCur = pcNxt; bCur = bNxt; labCur = labNxt;
    Nbound = (Nbound < S) ? Nbound : S;   // next N = V <= min(N, S)
  }
}